// Edge_atten_13219909337799
// MI455X (gfx1250) — compile-verified
//
#include <hip/hip_runtime.h>
#include <hip/hip_bf16.h>

typedef __attribute__((ext_vector_type(2))) float v2f;
typedef __attribute__((ext_vector_type(4))) float v4f;
typedef __attribute__((ext_vector_type(8))) float v8f;

#define LEAKY_ALPHA 0.2f

static constexpr int N_NODES = 50000;   // 3125 * 16
static constexpr int N_EDGES = 800000;  // 3125 * 256
static constexpr int EDGE_BLOCKS = 3125;
static constexpr int NODE_TILES = 3125; // N_NODES / 16

// Workspace layout (float offsets)
static constexpr size_t WS_A2   = 0;                                   // 16x128 folded weight (rows 8..15 zero)
static constexpr size_t WS_BIAS = 2048;                                // 8 folded biases
static constexpr size_t WS_S    = 2056;                                // [N,8] node scores (16B aligned)
static constexpr size_t WS_LOG  = WS_S + (size_t)N_NODES * 8;          // [E,4] logits
static constexpr size_t WS_MAXP = WS_LOG + (size_t)N_EDGES * 4;        // [3125,4] block maxes
static constexpr size_t WS_SUMP = WS_MAXP + (size_t)EDGE_BLOCKS * 4;   // [3125,4] block sums
static constexpr size_t WS_GMAX = WS_SUMP + (size_t)EDGE_BLOCKS * 4;   // [4]
static constexpr size_t WS_RSUM = WS_GMAX + 4;                         // [4]

// ---------------------------------------------------------------------------
// K0: fold attention vector a into W:  A2[j][c] = sum_k W[h*32+k][c]*a[h][half*32+k]
//     j = 2h + half  (half 0 = src part of a, half 1 = dst part). Rows 8..15 = 0.
// ---------------------------------------------------------------------------
__global__ void prep_kernel(const float* __restrict__ W, const float* __restrict__ b,
                            const float* __restrict__ a, float* ws) {
  const int c = threadIdx.x;  // 0..127
  float* A2 = ws + WS_A2;
#pragma unroll
  for (int j = 0; j < 8; ++j) {
    const int h = j >> 1, half = j & 1;
    const float* av = a + h * 64 + half * 32;
    float acc = 0.f;
#pragma unroll 8
    for (int k = 0; k < 32; ++k) acc += W[(size_t)(h * 32 + k) * 128 + c] * av[k];
    A2[j * 128 + c] = acc;
  }
#pragma unroll
  for (int j = 8; j < 16; ++j) A2[j * 128 + c] = 0.f;  // zero-pad B columns 8..15
  if (c < 8) {
    const int h = c >> 1, half = c & 1;
    const float* av = a + h * 64 + half * 32;
    float acc = 0.f;
    for (int k = 0; k < 32; ++k) acc += b[h * 32 + k] * av[k];
    ws[WS_BIAS + c] = acc;
  }
}

// ---------------------------------------------------------------------------
// K1: skinny GEMM s[N,8] = X[N,128] * A2^T via V_WMMA_F32_16X16X4_F32.
// One wave per 16-row tile; 32 WMMA steps over K=128.
// A layout (16x4 f32): lane = M (mod 16); VGPR0/1 = K pair {2*hi, 2*hi+1} + kbase.
// B layout (4x16 f32): lane = N (mod 16); VGPR0/1 = K pair {2*hi, 2*hi+1} + kbase.
// D layout (16x16 f32): VGPR r -> row r + 8*hi, col = lane mod 16.
// ---------------------------------------------------------------------------
__global__ void __launch_bounds__(256) score_wmma_kernel(const float* __restrict__ X,
                                                         float* ws) {
  const float* A2 = ws + WS_A2;
  const float* bias = ws + WS_BIAS;
  float* S = ws + WS_S;

  const int lane = threadIdx.x & 31;
  const int wave = threadIdx.x >> 5;
  const int tile = blockIdx.x * 8 + wave;
  if (tile >= NODE_TILES) return;  // wave-uniform: EXEC all-ones inside

  const int m = lane & 15;
  const int hi = lane >> 4;
  const float* xrow = X + (size_t)(tile * 16 + m) * 128 + 2 * hi;  // A operand
  const float* brow = A2 + (size_t)m * 128 + 2 * hi;               // B operand (col n = m)

  v8f acc = {};
#pragma unroll 8
  for (int kk = 0; kk < 32; ++kk) {
    v2f av = *(const v2f*)(xrow + kk * 4);
    v2f bv = *(const v2f*)(brow + kk * 4);
    acc = __builtin_amdgcn_wmma_f32_16x16x4_f32(
        /*neg_a=*/false, av, /*neg_b=*/false, bv,
        /*c_mod=*/(short)0, acc, /*reuse_a=*/false, /*reuse_b=*/false);
  }

  if (m < 8) {  // only columns 0..7 are real outputs
    const float bb = bias[m];
    const int rowbase = tile * 16 + 8 * hi;
#pragma unroll
    for (int r = 0; r < 8; ++r) S[(size_t)(rowbase + r) * 8 + m] = acc[r] + bb;
  }
}

// ---------------------------------------------------------------------------
// K2: per-edge logits + LeakyReLU; store logits; block-level max partials.
// ---------------------------------------------------------------------------
__global__ void __launch_bounds__(256) edge_logits_kernel(const int* __restrict__ ei,
                                                          float* ws) {
  __shared__ v4f red[256];
  const int t = threadIdx.x;
  const int e = blockIdx.x * 256 + t;
  const v4f* S = (const v4f*)(ws + WS_S);

  const int src = ei[e];
  const int dst = ei[N_EDGES + e];
  v4f s0 = S[(size_t)src * 2], s1 = S[(size_t)src * 2 + 1];
  v4f d0 = S[(size_t)dst * 2], d1 = S[(size_t)dst * 2 + 1];

  v4f lg;
  lg.x = s0.x + d0.y;  // head 0: s_src(j=0) + s_dst(j=1)
  lg.y = s0.z + d0.w;  // head 1
  lg.z = s1.x + d1.y;  // head 2
  lg.w = s1.z + d1.w;  // head 3
  lg.x = fmaxf(lg.x, LEAKY_ALPHA * lg.x);
  lg.y = fmaxf(lg.y, LEAKY_ALPHA * lg.y);
  lg.z = fmaxf(lg.z, LEAKY_ALPHA * lg.z);
  lg.w = fmaxf(lg.w, LEAKY_ALPHA * lg.w);

  *(v4f*)(ws + WS_LOG + (size_t)e * 4) = lg;

  red[t] = lg;
  __syncthreads();
  for (int off = 128; off > 0; off >>= 1) {
    if (t < off) {
      v4f o = red[t + off], m = red[t];
      m.x = fmaxf(m.x, o.x); m.y = fmaxf(m.y, o.y);
      m.z = fmaxf(m.z, o.z); m.w = fmaxf(m.w, o.w);
      red[t] = m;
    }
    __syncthreads();
  }
  if (t == 0) *(v4f*)(ws + WS_MAXP + (size_t)blockIdx.x * 4) = red[0];
}

// ---------------------------------------------------------------------------
// K3: deterministic global max over block partials.
// ---------------------------------------------------------------------------
__global__ void __launch_bounds__(256) reduce_max_kernel(float* ws) {
  __shared__ v4f red[256];
  const int t = threadIdx.x;
  const v4f* P = (const v4f*)(ws + WS_MAXP);
  v4f m = {-3.0e38f, -3.0e38f, -3.0e38f, -3.0e38f};
  for (int i = t; i < EDGE_BLOCKS; i += 256) {
    v4f o = P[i];
    m.x = fmaxf(m.x, o.x); m.y = fmaxf(m.y, o.y);
    m.z = fmaxf(m.z, o.z); m.w = fmaxf(m.w, o.w);
  }
  red[t] = m;
  __syncthreads();
  for (int off = 128; off > 0; off >>= 1) {
    if (t < off) {
      v4f o = red[t + off], q = red[t];
      q.x = fmaxf(q.x, o.x); q.y = fmaxf(q.y, o.y);
      q.z = fmaxf(q.z, o.z); q.w = fmaxf(q.w, o.w);
      red[t] = q;
    }
    __syncthreads();
  }
  if (t == 0) *(v4f*)(ws + WS_GMAX) = red[0];
}

// ---------------------------------------------------------------------------
// K4: exp(logit - gmax) -> out; block-level sum partials (fixed-order tree).
// ---------------------------------------------------------------------------
__global__ void __launch_bounds__(256) edge_exp_kernel(float* ws, float* __restrict__ out) {
  __shared__ v4f red[256];
  const int t = threadIdx.x;
  const int e = blockIdx.x * 256 + t;
  const v4f gm = *(const v4f*)(ws + WS_GMAX);
  v4f lg = *(const v4f*)(ws + WS_LOG + (size_t)e * 4);
  v4f ex;
  ex.x = __expf(lg.x - gm.x);
  ex.y = __expf(lg.y - gm.y);
  ex.z = __expf(lg.z - gm.z);
  ex.w = __expf(lg.w - gm.w);
  *(v4f*)(out + (size_t)e * 4) = ex;

  red[t] = ex;
  __syncthreads();
  for (int off = 128; off > 0; off >>= 1) {
    if (t < off) {
      v4f o = red[t + off], q = red[t];
      q.x += o.x; q.y += o.y; q.z += o.z; q.w += o.w;
      red[t] = q;
    }
    __syncthreads();
  }
  if (t == 0) *(v4f*)(ws + WS_SUMP + (size_t)blockIdx.x * 4) = red[0];
}

// ---------------------------------------------------------------------------
// K5: deterministic global sum over block partials -> reciprocal.
// ---------------------------------------------------------------------------
__global__ void __launch_bounds__(256) reduce_sum_kernel(float* ws) {
  __shared__ v4f red[256];
  const int t = threadIdx.x;
  const v4f* P = (const v4f*)(ws + WS_SUMP);
  v4f s = {};
  for (int i = t; i < EDGE_BLOCKS; i += 256) {  // fixed per-thread order
    v4f o = P[i];
    s.x += o.x; s.y += o.y; s.z += o.z; s.w += o.w;
  }
  red[t] = s;
  __syncthreads();
  for (int off = 128; off > 0; off >>= 1) {
    if (t < off) {
      v4f o = red[t + off], q = red[t];
      q.x += o.x; q.y += o.y; q.z += o.z; q.w += o.w;
      red[t] = q;
    }
    __syncthreads();
  }
  if (t == 0) {
    v4f r = red[0];
    v4f inv = {1.0f / r.x, 1.0f / r.y, 1.0f / r.z, 1.0f / r.w};
    *(v4f*)(ws + WS_RSUM) = inv;
  }
}

// ---------------------------------------------------------------------------
// K6: scale out by 1/sum per head.
// ---------------------------------------------------------------------------
__global__ void __launch_bounds__(256) scale_kernel(float* ws, float* __restrict__ out) {
  const int e = blockIdx.x * 256 + threadIdx.x;
  const v4f r = *(const v4f*)(ws + WS_RSUM);
  v4f v = *(v4f*)(out + (size_t)e * 4);
  v.x *= r.x; v.y *= r.y; v.z *= r.z; v.w *= r.w;
  *(v4f*)(out + (size_t)e * 4) = v;
}

extern "C" void kernel_launch(void* const* d_in, const int* in_sizes, int n_in,
                              void* d_out, int out_size, void* d_ws, size_t ws_size,
                              hipStream_t stream) {
  const float* node_feats = (const float*)d_in[0];
  const int*   edge_index = (const int*)d_in[1];  // JAX x64-off: int64 -> int32
  const float* W          = (const float*)d_in[2];
  const float* b          = (const float*)d_in[3];
  const float* a          = (const float*)d_in[4];
  float* out = (float*)d_out;
  float* ws  = (float*)d_ws;

  prep_kernel<<<1, 128, 0, stream>>>(W, b, a, ws);
  score_wmma_kernel<<<(NODE_TILES + 7) / 8, 256, 0, stream>>>(node_feats, ws);
  edge_logits_kernel<<<EDGE_BLOCKS, 256, 0, stream>>>(edge_index, ws);
  reduce_max_kernel<<<1, 256, 0, stream>>>(ws);
  edge_exp_kernel<<<EDGE_BLOCKS, 256, 0, stream>>>(ws, out);
  reduce_sum_kernel<<<1, 256, 0, stream>>>(ws);
  scale_kernel<<<EDGE_BLOCKS, 256, 0, stream>>>(ws, out);
}